// Model1_HSinner_7756710937246
// MI455X (gfx1250) — compile-verified
//
#include <hip/hip_runtime.h>
#include <hip/hip_bf16.h>

typedef __attribute__((ext_vector_type(2))) float v2f;
typedef __attribute__((ext_vector_type(8))) float v8f;

#define PI_F 3.14159265358979323846f

// ----------------------------------------------------------------------------
// ±1 feature-map matrix element for basis state n (0..255), feature k (0..14):
//   k in [0,8):  s_{n,k}   = +1 if bit (7-k) of n is 0, else -1   (wire0 = MSB)
//   k in [8,15): ss_{n,k-8} = s_{n,j}*s_{n,j+1}, j = k-8
// ----------------------------------------------------------------------------
__device__ __forceinline__ float sval(int k, int n) {
    bool zz = (k >= 8);
    int j = zz ? (k - 8) : k;        // j in [0,7] (k<=14 -> j<=6 when zz)
    int bit = (n >> (7 - j)) & 1;
    if (zz) bit ^= (n >> (6 - j)) & 1;
    return bit ? -1.0f : 1.0f;
}

// MLP 8 -> 10 -> 10 -> 8, weights packed in LDS:
//  W1[0..80) b1[80..90) W2[90..190) b2[190..200) W3[200..280) b3[280..288)
__device__ __forceinline__ void mlp8(const float* __restrict__ x,
                                     const float* w, float* f) {
    float xin[8];
#pragma unroll
    for (int j = 0; j < 8; ++j) xin[j] = x[j];
    float h1[10];
#pragma unroll
    for (int i = 0; i < 10; ++i) {
        float a = w[80 + i];
#pragma unroll
        for (int j = 0; j < 8; ++j) a = fmaf(xin[j], w[i * 8 + j], a);
        h1[i] = fmaxf(a, 0.0f);
    }
    float h2[10];
#pragma unroll
    for (int i = 0; i < 10; ++i) {
        float a = w[190 + i];
#pragma unroll
        for (int j = 0; j < 10; ++j) a = fmaf(h1[j], w[90 + i * 10 + j], a);
        h2[i] = fmaxf(a, 0.0f);
    }
#pragma unroll
    for (int o = 0; o < 8; ++o) {
        float a = w[280 + o];
#pragma unroll
        for (int j = 0; j < 10; ++j) a = fmaf(h2[j], w[200 + o * 10 + j], a);
        f[o] = a;
    }
}

// One wave handles 16 samples; 256-thread block = 8 waves = 128 samples.
__global__ __launch_bounds__(256) void hsinner_kernel(
    const float* __restrict__ x1, const float* __restrict__ x2,
    const float* __restrict__ W1, const float* __restrict__ b1,
    const float* __restrict__ W2, const float* __restrict__ b2,
    const float* __restrict__ W3, const float* __restrict__ b3,
    float* __restrict__ out, int nBatch) {
    __shared__ float w[288];
    for (int i = threadIdx.x; i < 288; i += blockDim.x) {
        float v;
        if (i < 80)       v = W1[i];
        else if (i < 90)  v = b1[i - 80];
        else if (i < 190) v = W2[i - 90];
        else if (i < 200) v = b2[i - 190];
        else if (i < 280) v = W3[i - 200];
        else              v = b3[i - 280];
        w[i] = v;
    }
    __syncthreads();

    const int wave = threadIdx.x >> 5;
    const int lane = threadIdx.x & 31;
    const int m = lane & 15;                       // sample row within wave tile
    const int sbase = (blockIdx.x * 8 + wave) * 16;
    if (sbase >= nBatch) return;                   // whole-wave uniform exit
    const int s = sbase + m;

    // --- feature vector: d[0..7], e[0..6], pad 0 ---------------------------
    float f1[8], f2[8];
    mlp8(x1 + (size_t)s * 8, w, f1);
    mlp8(x2 + (size_t)s * 8, w, f2);

    float feat[16];
#pragma unroll
    for (int j = 0; j < 8; ++j) feat[j] = f1[j] - f2[j];
#pragma unroll
    for (int j = 0; j < 7; ++j)
        feat[8 + j] = (PI_F - f1[j]) * (PI_F - f1[j + 1])
                    - (PI_F - f2[j]) * (PI_F - f2[j + 1]);
    feat[15] = 0.0f;

    // --- A fragments (16x4 f32, 2 VGPRs per K-chunk) -----------------------
    // chunk c, VGPR0: lanes 0-15 -> K=4c, lanes 16-31 -> K=4c+2; VGPR1: +1.
    const int koff = (lane >> 4) << 1;             // 0 or 2
    v2f A[4];
#pragma unroll
    for (int c = 0; c < 4; ++c) {
        A[c].x = (koff == 0) ? feat[4 * c + 0] : feat[4 * c + 2];
        A[c].y = (koff == 0) ? feat[4 * c + 1] : feat[4 * c + 3];
    }

    float sums[8];
#pragma unroll
    for (int r = 0; r < 8; ++r) sums[r] = 0.0f;

    // --- 16 N-tiles of 16 basis states each --------------------------------
#pragma unroll 1
    for (int nt = 0; nt < 16; ++nt) {
        const int n = nt * 16 + (lane & 15);
        v8f acc = {};
#pragma unroll
        for (int c = 0; c < 4; ++c) {
            const int k0 = 4 * c + koff;
            v2f Bf;
            Bf.x = sval(k0, n);                          // k0 <= 14 always
            Bf.y = (k0 + 1 == 15) ? 0.0f : sval(k0 + 1, n);
            acc = __builtin_amdgcn_wmma_f32_16x16x4_f32(
                false, A[c], false, Bf, (short)0, acc, false, false);
        }
#pragma unroll
        for (int r = 0; r < 8; ++r) sums[r] += __cosf(acc[r]);
    }

    // --- row reduction within 16-lane groups, then scatter -----------------
    // C/D layout: VGPR r, lanes 0-15 -> (M=r, N=lane); lanes 16-31 -> (M=8+r).
#pragma unroll
    for (int r = 0; r < 8; ++r) {
        float v = sums[r];
        v += __shfl_xor(v, 1, 32);
        v += __shfl_xor(v, 2, 32);
        v += __shfl_xor(v, 4, 32);
        v += __shfl_xor(v, 8, 32);
        if (lane == 0)  out[sbase + r]     = v * (1.0f / 256.0f);
        if (lane == 16) out[sbase + 8 + r] = v * (1.0f / 256.0f);
    }
}

extern "C" void kernel_launch(void* const* d_in, const int* in_sizes, int n_in,
                              void* d_out, int out_size, void* d_ws, size_t ws_size,
                              hipStream_t stream) {
    const float* x1 = (const float*)d_in[0];
    const float* x2 = (const float*)d_in[1];
    const float* W1 = (const float*)d_in[2];
    const float* b1 = (const float*)d_in[3];
    const float* W2 = (const float*)d_in[4];
    const float* b2 = (const float*)d_in[5];
    const float* W3 = (const float*)d_in[6];
    const float* b3 = (const float*)d_in[7];
    float* out = (float*)d_out;

    const int nBatch = in_sizes[0] / 8;            // 65536
    const int blocks = (nBatch + 127) / 128;       // 128 samples per block
    hsinner_kernel<<<blocks, 256, 0, stream>>>(x1, x2, W1, b1, W2, b2, W3, b3,
                                               out, nBatch);
}